// BlockSparseLinear_86182813761997
// MI455X (gfx1250) — compile-verified
//
#include <hip/hip_runtime.h>

typedef float v2f  __attribute__((ext_vector_type(2)));
typedef float v8f  __attribute__((ext_vector_type(8)));
typedef unsigned int u32x4 __attribute__((ext_vector_type(4)));
typedef int   i32x4 __attribute__((ext_vector_type(4)));
typedef int   i32x8 __attribute__((ext_vector_type(8)));

#define TOKENS 8192
#define IN_F   4096
#define OUT_F  4096
#define BS     32
#define NBR    (OUT_F / BS)   // 128 block rows
#define NBC    (IN_F / BS)    // 128 block cols
#define TM     128            // tokens per workgroup tile
#define LDSS   36             // padded LDS row stride (floats) == 32 + 4 (TDM pad)

// ---------------------------------------------------------------------------
// Pass 1a: zero the dense presence map (NBR*NBC ints in workspace).
// ---------------------------------------------------------------------------
__global__ void bsl_zero_map(int* __restrict__ map) {
    int i = blockIdx.x * blockDim.x + threadIdx.x;
    if (i < NBR * NBC) map[i] = 0;
}

// ---------------------------------------------------------------------------
// Pass 1b: scatter block ids. Positions unique -> no atomics, deterministic.
// ---------------------------------------------------------------------------
__global__ void bsl_build_map(const int* __restrict__ brow,
                              const int* __restrict__ bcol,
                              int* __restrict__ map, int nblk) {
    int i = blockIdx.x * blockDim.x + threadIdx.x;
    if (i < nblk) map[brow[i] * NBC + bcol[i]] = i + 1;   // 0 == empty
}

// ---------------------------------------------------------------------------
// Pass 1c: compact each row's active columns (in column order -> deterministic
// summation order). Tiny serial scan, runs once; negligible cost.
// ---------------------------------------------------------------------------
__global__ void bsl_compact(const int* __restrict__ map,
                            int* __restrict__ list, int* __restrict__ counts) {
    const int r = blockIdx.x;
    if (threadIdx.x == 0) {
        int n = 0;
        for (int c = 0; c < NBC; ++c) {
            int e = map[r * NBC + c];
            if (e) list[r * NBC + n++] = ((e - 1) << 8) | c;   // blk<<8 | col
        }
        counts[r] = n;
    }
}

// ---------------------------------------------------------------------------
// TDM 2D tile load: global (row-major, stride in elements) -> padded LDS.
// D# packing per CDNA5 ISA ch.8. data_size=4B; pad_interval=4 (32 DWORDs),
// pad_amount=3 (4 DWORDs) => LDS row stride of 36 floats (== LDSS).
// Toolchain: clang-23 6-arg builtin (g0, g1, g2, g3, g4, cpol).
// ---------------------------------------------------------------------------
__device__ __forceinline__ void tdm_load_2d(unsigned int lds_addr,
                                            const void* gptr,
                                            unsigned int tensor_d0,
                                            unsigned int tensor_d1,
                                            unsigned int tile_d0,
                                            unsigned int tile_d1,
                                            unsigned long long stride0) {
    const unsigned long long ga = (unsigned long long)(uintptr_t)gptr;
    u32x4 g0;
    g0[0] = 1u;                                            // count=1, user D#
    g0[1] = lds_addr;                                      // LDS byte address
    g0[2] = (unsigned int)ga;                              // global_addr[31:0]
    g0[3] = (unsigned int)((ga >> 32) & 0x01FFFFFFu)       // global_addr[56:32]
          | (2u << 30);                                    // type = 2 (image)
    i32x8 g1;
    g1[0] = (int)((2u << 16)                               // data_size = 4B
          |  (1u << 20)                                    // pad_enable
          |  (4u << 22)                                    // pad_interval: 32 DW
          |  (3u << 25));                                  // pad_amount: 4 DW
    g1[1] = (int)((tensor_d0 & 0xFFFFu) << 16);            // dim0[15:0] @ b63:48
    g1[2] = (int)((tensor_d0 >> 16) | ((tensor_d1 & 0xFFFFu) << 16));
    g1[3] = (int)(((tensor_d1 >> 16) & 0xFFFFu) | (tile_d0 << 16));
    g1[4] = (int)(tile_d1 & 0xFFFFu);                      // tile_dim2 = 0 (2D)
    g1[5] = (int)(unsigned int)stride0;                    // dim0_stride[31:0]
    g1[6] = (int)(unsigned int)((stride0 >> 32) & 0xFFFFu);
    g1[7] = 0;                                             // dim1_stride unused
    const i32x4 z4 = {0, 0, 0, 0};
    const i32x8 z8 = {0, 0, 0, 0, 0, 0, 0, 0};
    __builtin_amdgcn_tensor_load_to_lds(g0, g1, z4, z4, z8, 0);
}

// ---------------------------------------------------------------------------
// Pass 2: block-sparse GEMM. grid = (TOKENS/TM, NBR), 256 threads (8 wave32).
// Wave w: tokens [16w,16w+16) x 32 outputs -> four 16x16 f32 accumulators
// (even/odd K phases to break WMMA->WMMA C-dependency stalls).
// TDM double-buffers the x-tile (128x32) and W-tile (32x32) into padded LDS.
// ---------------------------------------------------------------------------
__global__ __launch_bounds__(256)
void bsl_gemm(const float* __restrict__ x,
              const float* __restrict__ wdata,
              const float* __restrict__ bias,
              const int* __restrict__ counts,
              const int* __restrict__ list,
              float* __restrict__ out) {
    __shared__ float pool[2 * TM * LDSS + 2 * BS * LDSS];  // ~46 KB
    const int tid   = threadIdx.x;
    const int lane  = tid & 31;
    const int wv    = tid >> 5;
    const int t0    = blockIdx.x * TM;
    const int r     = blockIdx.y;
    const int obase = r * BS;
    const int nb    = counts[r];

    // Flat LDS address truncates to the 32-bit LDS byte offset (ISA 10.2).
    const unsigned int lds_base = (unsigned int)(uintptr_t)(void*)pool;

    const int tbase = wv * 16;
    const int mlo   = lane & 15;          // M (A) / N (B,C) low index
    const int khalf = (lane >> 4) << 1;   // lane-half K offset: 0 or 2

    v8f acc00 = {}, acc01 = {}, acc10 = {}, acc11 = {};

    auto issue = [&](int ib) {            // wave0 only: DMA block ib -> buf ib&1
        const int e   = list[r * NBC + ib];
        const int blk = e >> 8;
        const int c   = e & 0xFF;
        const int b   = ib & 1;
        tdm_load_2d(lds_base + (unsigned int)(b * TM * LDSS) * 4u,
                    x + (size_t)t0 * IN_F + (size_t)c * BS,
                    IN_F, TOKENS, BS, TM, (unsigned long long)IN_F);
        tdm_load_2d(lds_base + (unsigned int)(2 * TM * LDSS + b * BS * LDSS) * 4u,
                    wdata + (size_t)blk * (BS * BS),
                    BS, BS, BS, BS, (unsigned long long)BS);
    };

    if (nb > 0) {
        if (wv == 0) issue(0);
        for (int ib = 0; ib < nb; ++ib) {
            __syncthreads();              // all waves done reading buf[(ib+1)&1]
            if (wv == 0) {
                if (ib + 1 < nb) {
                    issue(ib + 1);        // prefetch next block's tiles
                    __builtin_amdgcn_s_wait_tensorcnt(2);  // retire current buf
                } else {
                    __builtin_amdgcn_s_wait_tensorcnt(0);
                }
            }
            __syncthreads();              // release buf[ib&1] to all waves

            const int b = ib & 1;
            const float* xb  = pool + b * TM * LDSS;
            const float* wb  = pool + 2 * TM * LDSS + b * BS * LDSS;
            const float* xp  = &xb[(tbase + mlo) * LDSS + khalf]; // A: x[m][k]
            const float* wp0 = &wb[mlo * LDSS + khalf];           // B: W[n][k]
            const float* wp1 = wp0 + 16 * LDSS;
#pragma unroll
            for (int kk = 0; kk < BS; kk += 8) {
                v2f a0, a1, b00, b01, b10, b11;
                a0.x  = xp[kk];      a0.y  = xp[kk + 1];
                b00.x = wp0[kk];     b00.y = wp0[kk + 1];
                b01.x = wp1[kk];     b01.y = wp1[kk + 1];
                a1.x  = xp[kk + 4];  a1.y  = xp[kk + 5];
                b10.x = wp0[kk + 4]; b10.y = wp0[kk + 5];
                b11.x = wp1[kk + 4]; b11.y = wp1[kk + 5];
                acc00 = __builtin_amdgcn_wmma_f32_16x16x4_f32(
                    false, a0, false, b00, (short)0, acc00, false, false);
                acc01 = __builtin_amdgcn_wmma_f32_16x16x4_f32(
                    false, a0, false, b01, (short)0, acc01, false, false);
                acc10 = __builtin_amdgcn_wmma_f32_16x16x4_f32(
                    false, a1, false, b10, (short)0, acc10, false, false);
                acc11 = __builtin_amdgcn_wmma_f32_16x16x4_f32(
                    false, a1, false, b11, (short)0, acc11, false, false);
            }
        }
    }

    // Epilogue: merge K phases, add bias, store. C/D layout: VGPR v ->
    // M = v (lanes 0-15) or v+8 (lanes 16-31); N = lane & 15 per half.
    const int   moff  = (lane >> 4) << 3;
    const float bias0 = bias[obase + mlo];
    const float bias1 = bias[obase + 16 + mlo];
#pragma unroll
    for (int v = 0; v < 8; ++v) {
        const size_t t = (size_t)(t0 + tbase + moff + v);
        out[t * OUT_F + obase + mlo]      = acc00[v] + acc10[v] + bias0;
        out[t * OUT_F + obase + 16 + mlo] = acc01[v] + acc11[v] + bias1;
    }
}

// ---------------------------------------------------------------------------
extern "C" void kernel_launch(void* const* d_in, const int* in_sizes, int n_in,
                              void* d_out, int out_size, void* d_ws, size_t ws_size,
                              hipStream_t stream) {
    const float* x     = (const float*)d_in[0];
    const float* wdata = (const float*)d_in[1];
    const float* bias  = (const float*)d_in[2];
    const int*   brow  = (const int*)d_in[3];
    const int*   bcol  = (const int*)d_in[4];
    float*       out   = (float*)d_out;
    const int    nblk  = in_sizes[3];

    int* map    = (int*)d_ws;             // NBR*NBC ints
    int* list   = map + NBR * NBC;        // NBR*NBC ints
    int* counts = list + NBR * NBC;       // NBR ints

    bsl_zero_map <<<(NBR * NBC + 255) / 256, 256, 0, stream>>>(map);
    bsl_build_map<<<(nblk + 255) / 256, 256, 0, stream>>>(brow, bcol, map, nblk);
    bsl_compact  <<<NBR, 32, 0, stream>>>(map, list, counts);
    bsl_gemm     <<<dim3(TOKENS / TM, NBR), 256, 0, stream>>>(x, wdata, bias,
                                                              counts, list, out);
    (void)n_in; (void)out_size; (void)ws_size;
}